// Loss_yolov1_37675453120994
// MI455X (gfx1250) — compile-verified
//
#include <hip/hip_runtime.h>
#include <hip/hip_bf16.h>

typedef float v2f __attribute__((ext_vector_type(2)));
typedef float v8f __attribute__((ext_vector_type(8)));

// ---------------------------------------------------------------------------
// Wave-wide (32-lane) sum using V_WMMA_F32_16X16X4_F32.
// A (16x4 f32, 2 VGPRs/lane): lane L<16 holds (M=L,K=0),(M=L,K=1);
// lane L>=16 holds (M=L-16,K=2),(M=L-16,K=3). We place x in slot0, 0 in slot1,
// so rowsum(m) = x[lane m] + x[lane m+16]. B = all ones -> D[m][n] = rowsum(m).
// Sum of a lane's 8 D VGPRs = sum of rows 0..7 (lanes<16) or 8..15 (lanes>=16);
// one shfl_xor(16) gives the full wave total in every lane.
// Requires full-wave EXEC (all call sites below are convergent).
// ---------------------------------------------------------------------------
__device__ __forceinline__ float wave_sum32(float x) {
    v2f a; a[0] = x;   a[1] = 0.0f;
    v2f b; b[0] = 1.0f; b[1] = 1.0f;
    v8f c = {};
    v8f d = __builtin_amdgcn_wmma_f32_16x16x4_f32(
        /*neg_a=*/false, a, /*neg_b=*/false, b,
        /*c_mod=*/(short)0, c, /*reuse_a=*/false, /*reuse_b=*/false);
    float s = d[0] + d[1] + d[2] + d[3] + d[4] + d[5] + d[6] + d[7];
    s += __shfl_xor(s, 16, 32);
    return s;
}

// Block-wide sum; every thread returns the block total. All WMMA executions
// happen with all 32 lanes of each wave active.
__device__ __forceinline__ float block_sum(float x, float* lds) {
    const int lane = threadIdx.x & 31;
    const int wid  = threadIdx.x >> 5;
    float w = wave_sum32(x);
    if (lane == 0) lds[wid] = w;
    __syncthreads();
    const int nw = blockDim.x >> 5;           // 8 for 256 threads
    float t = (lane < nw) ? lds[lane] : 0.0f; // data-select only; no EXEC split
    return wave_sum32(t);
}

__device__ __forceinline__ float iou_xyxy(float x11, float y11, float x12, float y12,
                                          float x21, float y21, float x22, float y22) {
    bool no_ov = (x12 < x21) || (x11 > x22) || (y12 < y21) || (y11 > y22);
    float ix1 = fmaxf(x11, x21);
    float iy1 = fmaxf(y11, y21);
    float ix2 = fminf(x12, x22);
    float iy2 = fminf(y12, y22);
    float inter = no_ov ? 0.0f : (ix2 - ix1) * (iy2 - iy1);
    float a1 = (x12 - x11) * (y12 - y11);
    float a2 = (x22 - x21) * (y22 - y21);
    return (inter > 0.0f) ? inter / (a1 + a2 - inter) : 0.0f;
}

// Kernel 1: per-block partial sums of the per-cell YOLO loss.
__global__ void yolo_loss_partials(const float* __restrict__ pred,
                                   const float* __restrict__ lab,
                                   float* __restrict__ partials,
                                   int ncells) {
    __shared__ float lds[32];
    const int stride = gridDim.x * blockDim.x;
    float acc = 0.0f;

    for (int i = blockIdx.x * blockDim.x + threadIdx.x; i < ncells; i += stride) {
        const int b = i / 49;
        const int r = i - b * 49;          // m*7 + n
        const int m = r / 7;
        const int n = r - m * 7;
        const int base = b * 833 + r;      // 833 = 17*49

        float p[17], l[17];
        #pragma unroll
        for (int c = 0; c < 17; ++c) p[c] = pred[base + 49 * c];
        #pragma unroll
        for (int c = 0; c < 17; ++c) l[c] = lab[base + 49 * c];

        const float fn = (float)n, fm = (float)m;

        // box 1 (pred ch 0..3)
        float cx1 = (p[0] + fn) / 7.0f, cy1 = (p[1] + fm) / 7.0f;
        float b1x1 = cx1 - 0.5f * p[2], b1y1 = cy1 - 0.5f * p[3];
        float b1x2 = cx1 + 0.5f * p[2], b1y2 = cy1 + 0.5f * p[3];
        // box 2 (pred ch 5..8)
        float cx2 = (p[5] + fn) / 7.0f, cy2 = (p[6] + fm) / 7.0f;
        float b2x1 = cx2 - 0.5f * p[7], b2y1 = cy2 - 0.5f * p[8];
        float b2x2 = cx2 + 0.5f * p[7], b2y2 = cy2 + 0.5f * p[8];
        // ground box (labels ch 0..3)
        float cxg = (l[0] + fn) / 7.0f, cyg = (l[1] + fm) / 7.0f;
        float bgx1 = cxg - 0.5f * l[2], bgy1 = cyg - 0.5f * l[3];
        float bgx2 = cxg + 0.5f * l[2], bgy2 = cyg + 0.5f * l[3];

        float iou1 = iou_xyxy(b1x1, b1y1, b1x2, b1y2, bgx1, bgy1, bgx2, bgy2);
        float iou2 = iou_xyxy(b2x1, b2y1, b2x2, b2y2, bgx1, bgy1, bgx2, bgy2);
        bool use1 = (iou1 >= iou2);

        float d0 = p[0] - l[0], d1 = p[1] - l[1];
        float s2 = sqrtf(p[2]) - sqrtf(l[2]);
        float s3 = sqrtf(p[3]) - sqrtf(l[3]);
        float coor1 = 5.0f * (d0 * d0 + d1 * d1 + s2 * s2 + s3 * s3);

        float d5 = p[5] - l[5], d6 = p[6] - l[6];
        float s7 = sqrtf(p[7]) - sqrtf(l[7]);
        float s8 = sqrtf(p[8]) - sqrtf(l[8]);
        float coor2 = 5.0f * (d5 * d5 + d6 * d6 + s7 * s7 + s8 * s8);

        float e1 = p[4] - iou1; e1 *= e1;
        float e2 = p[9] - iou2; e2 *= e2;

        float coor          = use1 ? coor1 : coor2;
        float obj_conf      = use1 ? e1 : e2;
        float noobj_of_obj  = use1 ? 0.5f * e2 : 0.5f * e1;

        float cls = 0.0f;
        #pragma unroll
        for (int c = 10; c < 17; ++c) {
            float dd = p[c] - l[c];
            cls += dd * dd;
        }

        float obj_mask = (l[4] == 1.0f) ? 1.0f : 0.0f;
        float noobj_plain = 0.5f * (p[4] * p[4] + p[9] * p[9]);

        acc += obj_mask * (coor + obj_conf + noobj_of_obj + cls)
             + (1.0f - obj_mask) * noobj_plain;
    }

    float tot = block_sum(acc, lds);
    if (threadIdx.x == 0) partials[blockIdx.x] = tot;
}

// Kernel 2: deterministic final reduction of block partials; writes mean.
__global__ void yolo_loss_finalize(const float* __restrict__ partials,
                                   int nparts,
                                   float* __restrict__ out,
                                   float inv_batch) {
    __shared__ float lds[32];
    float acc = 0.0f;
    for (int i = threadIdx.x; i < nparts; i += blockDim.x) acc += partials[i];
    float tot = block_sum(acc, lds);
    if (threadIdx.x == 0) out[0] = tot * inv_batch;
}

extern "C" void kernel_launch(void* const* d_in, const int* in_sizes, int n_in,
                              void* d_out, int out_size, void* d_ws, size_t ws_size,
                              hipStream_t stream) {
    (void)n_in; (void)out_size;
    const float* pred = (const float*)d_in[0];
    const float* lab  = (const float*)d_in[1];
    float* out      = (float*)d_out;
    float* partials = (float*)d_ws;

    const int B      = in_sizes[0] / 833;   // 17*7*7 floats per batch item
    const int ncells = B * 49;

    const int block = 256;
    int nblocks = (ncells + block - 1) / block;     // 3136 for B=16384
    if (nblocks > 4096) nblocks = 4096;
    const int max_parts = (int)(ws_size / sizeof(float));
    if (nblocks > max_parts) nblocks = max_parts;
    if (nblocks < 1) nblocks = 1;

    yolo_loss_partials<<<nblocks, block, 0, stream>>>(pred, lab, partials, ncells);
    yolo_loss_finalize<<<1, block, 0, stream>>>(partials, nblocks, out, 1.0f / (float)B);
}